// LSTMDiscriminator_60756607369781
// MI455X (gfx1250) — compile-verified
//
#include <hip/hip_runtime.h>
#include <math.h>

// CDNA5 full-precision WMMA: D(16x16,f32) = A(16x4,f32) x B(4x16,f32) + C
typedef __attribute__((ext_vector_type(2))) float v2f;
typedef __attribute__((ext_vector_type(8))) float v8f;

#define WMMA_F32_16x16x4(a, b, c) \
    __builtin_amdgcn_wmma_f32_16x16x4_f32(false, (a), false, (b), (short)0, (c), false, false)

// Problem constants (hard-coded in the reference module)
#define BB 64        // batch
#define DD 500       // hidden width
#define DPAD 512     // padded row stride for intermediates
#define TT 2048      // timesteps (only the last one matters)

// ---------------------------------------------------------------------------
// Stage 1: add = P@Wp^T + bp + V@Wv^T + bv + dur*Wd^T + bd + step*Ws^T + bs
// One wave per 16x16 output tile. grid = (32 col tiles, 4 row tiles).
// Two independent WMMA accumulator chains (pitches / velocities) interleaved.
//
// Out-of-range columns (n >= 500): loads are clamped to row 0 (in-bounds);
// the garbage only lands in output columns we never store, because WMMA
// output column N depends solely on B column N (lanes with lane%16 == N).
// ---------------------------------------------------------------------------
__global__ __launch_bounds__(32)
void embed_kernel(const float* __restrict__ P,   // [64][128] (last timestep)
                  const float* __restrict__ V,   // [64][128]
                  const float* __restrict__ Dur, // [64]
                  const float* __restrict__ Stp, // [64]
                  const float* __restrict__ Wp, const float* __restrict__ bp,
                  const float* __restrict__ Wv, const float* __restrict__ bv,
                  const float* __restrict__ Wd, const float* __restrict__ bd,
                  const float* __restrict__ Ws, const float* __restrict__ bs,
                  float* __restrict__ out)       // [64][DPAD]
{
    const int lane = threadIdx.x & 31;
    const int half = lane >> 4;        // which half-wave (selects K sub-pair)
    const int l16  = lane & 15;
    const int colBase = blockIdx.x * 16;
    const int rowBase = blockIdx.y * 16;
    const int n = colBase + l16;                 // output feature
    const bool nvalid = (n < DD);
    const int nc = nvalid ? n : 0;               // clamped (keeps loads in-bounds)

    const float* arowP = P + (size_t)(rowBase + l16) * 128 + half * 2;
    const float* arowV = V + (size_t)(rowBase + l16) * 128 + half * 2;
    const float* browP = Wp + (size_t)nc * 128 + half * 2;
    const float* browV = Wv + (size_t)nc * 128 + half * 2;

    v8f acc0 = {};   // pitches @ Wp^T
    v8f acc1 = {};   // velocities @ Wv^T
    for (int k = 0; k < 128; k += 4) {
        v2f a0 = *(const v2f*)(arowP + k);
        v2f b0 = *(const v2f*)(browP + k);
        v2f a1 = *(const v2f*)(arowV + k);
        v2f b1 = *(const v2f*)(browV + k);
        acc0 = WMMA_F32_16x16x4(a0, b0, acc0);
        acc1 = WMMA_F32_16x16x4(a1, b1, acc1);
    }

    // rank-1 terms + biases (per-column values)
    if (nvalid) {
        const float bsum = bp[n] + bv[n] + bd[n] + bs[n];
        const float wdn  = Wd[n];   // Wd is [500][1]
        const float wsn  = Ws[n];
        #pragma unroll
        for (int r = 0; r < 8; ++r) {
            const int m = rowBase + r + half * 8;
            out[(size_t)m * DPAD + n] =
                acc0[r] + acc1[r] + bsum + Dur[m] * wdn + Stp[m] * wsn;
        }
    }
}

// ---------------------------------------------------------------------------
// Stage 2/3/4: out = act( X @ W^T + bias ),  X:[64][DPAD] (valid cols < 500),
// W:[500][500].  K = 500 = 125 WMMA steps; split across two accumulator
// chains (62 even/odd pairs + 1 tail) to hide WMMA latency.
// ---------------------------------------------------------------------------
template <bool LRELU>
__global__ __launch_bounds__(32)
void layer_kernel(const float* __restrict__ X,
                  const float* __restrict__ W,
                  const float* __restrict__ bias,
                  float* __restrict__ out)       // [64][DPAD]
{
    const int lane = threadIdx.x & 31;
    const int half = lane >> 4;
    const int l16  = lane & 15;
    const int colBase = blockIdx.x * 16;
    const int rowBase = blockIdx.y * 16;
    const int n = colBase + l16;
    const bool nvalid = (n < DD);
    const int nc = nvalid ? n : 0;

    const float* arow = X + (size_t)(rowBase + l16) * DPAD + half * 2;
    const float* brow = W + (size_t)nc * DD + half * 2;

    v8f acc0 = {};
    v8f acc1 = {};
    int k = 0;
    for (; k < 496; k += 8) {                    // 62 pairs
        v2f a0 = *(const v2f*)(arow + k);
        v2f b0 = *(const v2f*)(brow + k);
        v2f a1 = *(const v2f*)(arow + k + 4);
        v2f b1 = *(const v2f*)(brow + k + 4);
        acc0 = WMMA_F32_16x16x4(a0, b0, acc0);
        acc1 = WMMA_F32_16x16x4(a1, b1, acc1);
    }
    {                                            // tail: k = 496
        v2f a0 = *(const v2f*)(arow + k);
        v2f b0 = *(const v2f*)(brow + k);
        acc0 = WMMA_F32_16x16x4(a0, b0, acc0);
    }

    if (nvalid) {
        const float bn = bias[n];
        #pragma unroll
        for (int r = 0; r < 8; ++r) {
            const int m = rowBase + r + half * 8;
            float v = acc0[r] + acc1[r] + bn;
            if (LRELU) v = (v > 0.f) ? v : 0.01f * v;
            out[(size_t)m * DPAD + n] = v;
        }
    }
}

// ---------------------------------------------------------------------------
// Stage 5: out[b] = sigmoid( info[b,:] . W3 + b3 )   — 64 dots of length 500
// ---------------------------------------------------------------------------
__global__ __launch_bounds__(64)
void final_kernel(const float* __restrict__ info, // [64][DPAD]
                  const float* __restrict__ W3,   // [1][500]
                  const float* __restrict__ b3,   // [1]
                  float* __restrict__ out)        // [64]
{
    const int b = threadIdx.x;
    if (b < BB) {
        const float* row = info + (size_t)b * DPAD;
        float s0 = 0.f, s1 = 0.f;
        for (int i = 0; i < DD; i += 2) {
            s0 = fmaf(row[i], W3[i], s0);
            if (i + 1 < DD) s1 = fmaf(row[i + 1], W3[i + 1], s1);
        }
        const float s = s0 + s1 + b3[0];
        out[b] = 1.0f / (1.0f + expf(-s));
    }
}

// ---------------------------------------------------------------------------
extern "C" void kernel_launch(void* const* d_in, const int* in_sizes, int n_in,
                              void* d_out, int out_size, void* d_ws, size_t ws_size,
                              hipStream_t stream) {
    (void)in_sizes; (void)n_in; (void)out_size; (void)ws_size;

    const float* pitches    = (const float*)d_in[0];
    const float* velocities = (const float*)d_in[1];
    const float* durations  = (const float*)d_in[2];
    const float* steps      = (const float*)d_in[3];
    const float* Wp = (const float*)d_in[4];  const float* bp = (const float*)d_in[5];
    const float* Wv = (const float*)d_in[6];  const float* bv = (const float*)d_in[7];
    const float* Wd = (const float*)d_in[8];  const float* bd = (const float*)d_in[9];
    const float* Ws = (const float*)d_in[10]; const float* bs = (const float*)d_in[11];
    const float* W1 = (const float*)d_in[12]; const float* b1 = (const float*)d_in[13];
    const float* W2 = (const float*)d_in[14]; const float* b2 = (const float*)d_in[15];
    const float* Wi = (const float*)d_in[16]; const float* bi = (const float*)d_in[17];
    const float* W3 = (const float*)d_in[18]; const float* b3 = (const float*)d_in[19];

    // Only the last timestep matters (see reference comment).
    const float* Plast = pitches    + (size_t)(TT - 1) * BB * 128;
    const float* Vlast = velocities + (size_t)(TT - 1) * BB * 128;
    const float* Dlast = durations  + (size_t)(TT - 1) * BB;
    const float* Slast = steps      + (size_t)(TT - 1) * BB;

    // Two ping-pong [64][512] fp32 buffers in workspace (256 KB total).
    float* buf0 = (float*)d_ws;
    float* buf1 = buf0 + (size_t)BB * DPAD;

    const dim3 grid(DPAD / 16, BB / 16);   // 32 col tiles x 4 row tiles
    const dim3 block(32);                  // one wave32 per tile

    embed_kernel<<<grid, block, 0, stream>>>(Plast, Vlast, Dlast, Slast,
                                             Wp, bp, Wv, bv, Wd, bd, Ws, bs, buf0);
    layer_kernel<true ><<<grid, block, 0, stream>>>(buf0, W1, b1, buf1);
    layer_kernel<true ><<<grid, block, 0, stream>>>(buf1, W2, b2, buf0);
    layer_kernel<false><<<grid, block, 0, stream>>>(buf0, Wi, bi, buf1);
    final_kernel<<<1, 64, 0, stream>>>(buf1, W3, b3, (float*)d_out);
}